// GreedyCTCDecoder_42863773614250
// MI455X (gfx1250) — compile-verified
//
#include <hip/hip_runtime.h>
#include <math.h>
#include <stdint.h>

// Greedy CTC decode for emission[T=65536, V=512] fp32.
//   out[0..T)   = argmax over labels per timestep (as float)
//   out[T..2T)  = keep mask after blank/repeat collapse (0.0 / 1.0)
//
// Memory-bound: 128 MB streamed once. One wave32 per row; rows staged into
// LDS with CDNA5 async global->LDS copies, double-buffered per wave.

#define CTC_T 65536
#define CTC_V 512
#define ROW_BYTES 2048          // 512 floats
#define ROWS_PER_WAVE 8
#define WAVES_PER_BLOCK 8
#define BLOCK_THREADS (WAVES_PER_BLOCK * 32)

// Issue 4 async b128 copies: lane covers 16B per chunk, wave covers 512B per
// instruction, 4 instructions cover the full 2KB row. IOFFSET is added to both
// the global address and the LDS address (CDNA5 ISA).
__device__ __forceinline__ void async_load_row(const float* __restrict__ rowPtr,
                                               unsigned ldsOff, int lane) {
  unsigned long long g =
      (unsigned long long)(uintptr_t)rowPtr + (unsigned)(lane * 16);
  unsigned l = ldsOff + (unsigned)(lane * 16);
  asm volatile(
      "global_load_async_to_lds_b128 %0, %1, off\n\t"
      "global_load_async_to_lds_b128 %0, %1, off offset:512\n\t"
      "global_load_async_to_lds_b128 %0, %1, off offset:1024\n\t"
      "global_load_async_to_lds_b128 %0, %1, off offset:1536"
      :
      : "v"(l), "v"(g)
      : "memory");
}

__global__ __launch_bounds__(BLOCK_THREADS) void ctc_argmax_kernel(
    const float* __restrict__ emission, float* __restrict__ out_index) {
  __shared__ __align__(16) unsigned char smem[WAVES_PER_BLOCK * 2 * ROW_BYTES];

  const int lane = threadIdx.x & 31;
  const int wid = threadIdx.x >> 5;
  const int gw = blockIdx.x * WAVES_PER_BLOCK + wid;  // global wave id
  const int row0 = gw * ROWS_PER_WAVE;

  // Low 32 bits of a generic pointer into __shared__ == LDS byte offset.
  const unsigned ldsWaveBase =
      (unsigned)(uintptr_t)(&smem[wid * 2 * ROW_BYTES]);

  // Prime the pipeline: row0 -> buffer 0.
  async_load_row(emission + (size_t)row0 * CTC_V, ldsWaveBase, lane);

#pragma unroll
  for (int r = 0; r < ROWS_PER_WAVE; ++r) {
    if (r + 1 < ROWS_PER_WAVE) {
      // Prefetch next row into the other buffer, then wait until only those
      // 4 new ops remain outstanding (in-order completion => current buffer
      // is fully resident).
      async_load_row(emission + (size_t)(row0 + r + 1) * CTC_V,
                     ldsWaveBase + (unsigned)(((r + 1) & 1) * ROW_BYTES), lane);
      asm volatile("s_wait_asynccnt 4" ::: "memory");
    } else {
      asm volatile("s_wait_asynccnt 0" ::: "memory");
    }

    const unsigned bufOff =
        (unsigned)(wid * 2 * ROW_BYTES + (r & 1) * ROW_BYTES);

    // Lane-local argmax over its 16 elements (strict > keeps first occurrence,
    // matching jnp.argmax tie-break; indices scanned in increasing order).
    float bv = -INFINITY;
    int bi = 0;
#pragma unroll
    for (int c = 0; c < 4; ++c) {
      const float4 v4 = *reinterpret_cast<const float4*>(
          &smem[bufOff + (unsigned)(c * 512 + lane * 16)]);
      const int base = c * 128 + lane * 4;
      if (v4.x > bv) { bv = v4.x; bi = base + 0; }
      if (v4.y > bv) { bv = v4.y; bi = base + 1; }
      if (v4.z > bv) { bv = v4.z; bi = base + 2; }
      if (v4.w > bv) { bv = v4.w; bi = base + 3; }
    }

    // Wave32 butterfly reduce: max value, lowest index on ties.
#pragma unroll
    for (int m = 16; m >= 1; m >>= 1) {
      const float ov = __shfl_xor(bv, m, 32);
      const int oi = __shfl_xor(bi, m, 32);
      if (ov > bv || (ov == bv && oi < bi)) {
        bv = ov;
        bi = oi;
      }
    }

    if (lane == 0) {
      out_index[row0 + r] = (float)bi;  // label ids < 512, exact in fp32
    }
  }
}

__global__ __launch_bounds__(256) void ctc_keep_kernel(float* __restrict__ out,
                                                       int T) {
  const int t = blockIdx.x * blockDim.x + threadIdx.x;
  if (t >= T) return;

  const int idx = (int)out[t];
  // blank(0) -> -1 ; i>1 -> i-1 ; i==1 -> 0
  const int ch = (idx == 0) ? -1 : ((idx > 1) ? idx - 1 : 0);

  int prev;
  if (t == 0) {
    prev = -2;
  } else {
    const int pidx = (int)out[t - 1];
    prev = (pidx == 0) ? -1 : ((pidx > 1) ? pidx - 1 : 0);
  }

  out[T + t] = ((ch != prev) && (ch != -1)) ? 1.0f : 0.0f;
}

extern "C" void kernel_launch(void* const* d_in, const int* in_sizes, int n_in,
                              void* d_out, int out_size, void* d_ws,
                              size_t ws_size, hipStream_t stream) {
  (void)in_sizes;
  (void)n_in;
  (void)d_ws;
  (void)ws_size;
  (void)out_size;

  const float* emission = (const float*)d_in[0];
  float* out = (float*)d_out;

  // 65536 rows / (8 waves/block * 8 rows/wave) = 1024 blocks.
  const int blocks = CTC_T / (WAVES_PER_BLOCK * ROWS_PER_WAVE);
  ctc_argmax_kernel<<<blocks, BLOCK_THREADS, 0, stream>>>(emission, out);

  const int keepBlocks = (CTC_T + 255) / 256;
  ctc_keep_kernel<<<keepBlocks, 256, 0, stream>>>(out, CTC_T);
}